// GeneratorNet_27487790695136
// MI455X (gfx1250) — compile-verified
//
#include <hip/hip_runtime.h>
#include <hip/hip_bf16.h>

typedef __attribute__((ext_vector_type(2))) float v2f;
typedef __attribute__((ext_vector_type(8))) float v8f;

#define WAY   32
#define QUERY 128
#define FEAT  640
#define NQ    (WAY * QUERY)        // 4096 queries
#define ROWS  (WAY * (WAY - 1))    // 992 pair rows

__device__ __forceinline__ float leaky(float x) { return x > 0.0f ? x : 0.2f * x; }

// ---------------------------------------------------------------------------
// K1: AB[32,1280] = S[32,640] @ [ W1[:, :640]^T | W1[:, 640:]^T ]
//     One wave (32 threads) per 16x16 tile; V_WMMA_F32_16X16X4_F32, K-step 4.
//     A frag (16x4):  lanes 0-15 hold K={k,k+1}, lanes 16-31 hold K={k+2,k+3}
//     B frag (4x16):  same split on K, N striped across lanes.
// ---------------------------------------------------------------------------
__global__ void __launch_bounds__(32) gemm_ab(const float* __restrict__ S,
                                              const float* __restrict__ W1,
                                              float* __restrict__ AB) {
    const int lane  = threadIdx.x;
    const int half  = lane >> 4;       // 0 | 1
    const int lrow  = lane & 15;
    const int tileN = blockIdx.x;      // 0..79  (1280 cols)
    const int tileM = blockIdx.y;      // 0..1   (32 rows)

    const int m = tileM * 16 + lrow;
    const int n = tileN * 16 + lrow;   // 0..1279: [0,640)->W1a slab, [640,1280)->W1b slab

    const float* __restrict__ arow = S + (size_t)m * FEAT;
    const float* __restrict__ brow = (n < FEAT)
        ? (W1 + (size_t)n * (2 * FEAT))                    // W1[n, k]
        : (W1 + (size_t)(n - FEAT) * (2 * FEAT) + FEAT);   // W1[n-640, k+640]

    v8f c = {};
#pragma unroll 4
    for (int k = 0; k < FEAT; k += 4) {
        const int kk = k + (half << 1);
        v2f a = *(const v2f*)(arow + kk);   // 8B aligned (kk even, row base 8B-mult)
        v2f b = *(const v2f*)(brow + kk);
        c = __builtin_amdgcn_wmma_f32_16x16x4_f32(
                /*neg_a=*/false, a, /*neg_b=*/false, b,
                /*c_mod=*/(short)0, c, /*reuse_a=*/false, /*reuse_b=*/false);
    }

    // C/D layout: VGPR v -> M = tileM*16 + v + 8*half, N = tileN*16 + lrow
    const int col = tileN * 16 + lrow;
#pragma unroll
    for (int v = 0; v < 8; ++v) {
        const int row = tileM * 16 + v + 8 * half;
        AB[(size_t)row * (2 * FEAT) + col] = c[v];
    }
}

// ---------------------------------------------------------------------------
// K2: X1[i*31+jj, o] = leaky( AB[i, o] + AB[j, 640+o] + b1[o] ),  j = jj + (jj>=i)
// ---------------------------------------------------------------------------
__global__ void build_x1(const float* __restrict__ AB,
                         const float* __restrict__ b1,
                         float* __restrict__ X1) {
    const int idx = blockIdx.x * blockDim.x + threadIdx.x;
    if (idx >= ROWS * FEAT) return;
    const int r = idx / FEAT;
    const int o = idx - r * FEAT;
    const int i  = r / (WAY - 1);
    const int jj = r - i * (WAY - 1);
    const int j  = jj + (jj >= i ? 1 : 0);
    const float v = AB[(size_t)i * (2 * FEAT) + o]
                  + AB[(size_t)j * (2 * FEAT) + FEAT + o]
                  + b1[o];
    X1[idx] = leaky(v);
}

// ---------------------------------------------------------------------------
// K3: X2[992,640] = leaky( X1[992,640] @ W2^T + b2 )   (WMMA f32, 62x40 tiles)
// ---------------------------------------------------------------------------
__global__ void __launch_bounds__(32) gemm_x2(const float* __restrict__ X1,
                                              const float* __restrict__ W2,
                                              const float* __restrict__ b2,
                                              float* __restrict__ X2) {
    const int lane  = threadIdx.x;
    const int half  = lane >> 4;
    const int lrow  = lane & 15;
    const int tileN = blockIdx.x;      // 0..39
    const int tileM = blockIdx.y;      // 0..61

    const int m = tileM * 16 + lrow;
    const int n = tileN * 16 + lrow;

    const float* __restrict__ arow = X1 + (size_t)m * FEAT;
    const float* __restrict__ brow = W2 + (size_t)n * FEAT;   // W2[n, k]

    v8f c = {};
#pragma unroll 4
    for (int k = 0; k < FEAT; k += 4) {
        const int kk = k + (half << 1);
        v2f a = *(const v2f*)(arow + kk);
        v2f b = *(const v2f*)(brow + kk);
        c = __builtin_amdgcn_wmma_f32_16x16x4_f32(
                false, a, false, b, (short)0, c, false, false);
    }

    const int col  = tileN * 16 + lrow;
    const float bb = b2[col];
#pragma unroll
    for (int v = 0; v < 8; ++v) {
        const int row = tileM * 16 + v + 8 * half;
        X2[(size_t)row * FEAT + col] = leaky(c[v] + bb);
    }
}

// ---------------------------------------------------------------------------
// K4: c2[i, o] = ( mean_jj X2[i*31+jj, o] )^2
// ---------------------------------------------------------------------------
__global__ void ctx_c2(const float* __restrict__ X2, float* __restrict__ C2) {
    const int idx = blockIdx.x * blockDim.x + threadIdx.x;
    if (idx >= WAY * FEAT) return;
    const int i = idx / FEAT;
    const int o = idx - i * FEAT;
    float s = 0.0f;
#pragma unroll 31
    for (int jj = 0; jj < WAY - 1; ++jj)
        s += X2[(size_t)(i * (WAY - 1) + jj) * FEAT + o];
    const float m = s * (1.0f / (WAY - 1));
    C2[idx] = m * m;
}

// ---------------------------------------------------------------------------
// K5: out[q, i] = -sqrt( sum_f ((S[i,f] - Q[q,f]) * c2[i,f])^2 )
//     One wave32 per (q,i); lane-strided f, shfl_xor tree reduction.
// ---------------------------------------------------------------------------
__global__ void __launch_bounds__(256) dist_kernel(const float* __restrict__ S,
                                                   const float* __restrict__ Q,
                                                   const float* __restrict__ C2,
                                                   float* __restrict__ out) {
    const int w    = blockIdx.x * (blockDim.x >> 5) + (threadIdx.x >> 5);
    const int lane = threadIdx.x & 31;
    const int q = w >> 5;     // 0..4095
    const int i = w & 31;     // 0..31

    const float* __restrict__ sp = S  + (size_t)i * FEAT;
    const float* __restrict__ cp = C2 + (size_t)i * FEAT;
    const float* __restrict__ qp = Q  + (size_t)q * FEAT;

    float acc = 0.0f;
#pragma unroll 5
    for (int f = lane; f < FEAT; f += 32) {
        const float d = (sp[f] - qp[f]) * cp[f];
        acc += d * d;
    }
#pragma unroll
    for (int off = 16; off > 0; off >>= 1)
        acc += __shfl_xor(acc, off, 32);

    if (lane == 0) out[(size_t)q * WAY + i] = -sqrtf(acc);
}

// ---------------------------------------------------------------------------
// Launch
// ---------------------------------------------------------------------------
extern "C" void kernel_launch(void* const* d_in, const int* in_sizes, int n_in,
                              void* d_out, int out_size, void* d_ws, size_t ws_size,
                              hipStream_t stream) {
    const float* S  = (const float*)d_in[0];   // [32, 640]
    const float* Q  = (const float*)d_in[1];   // [32, 128, 640] -> [4096, 640]
    const float* W1 = (const float*)d_in[2];   // [640, 1280]
    const float* b1 = (const float*)d_in[3];   // [640]
    const float* W2 = (const float*)d_in[4];   // [640, 640]
    const float* b2 = (const float*)d_in[5];   // [640]
    float* out = (float*)d_out;                // [4096, 32]

    float* ws = (float*)d_ws;
    float* AB = ws;                            // 32 * 1280    = 40,960 floats
    float* X1 = AB + (size_t)WAY * 2 * FEAT;   // 992 * 640    = 634,880 floats
    float* X2 = X1 + (size_t)ROWS * FEAT;      // 992 * 640    = 634,880 floats
    float* C2 = X2 + (size_t)ROWS * FEAT;      // 32 * 640     = 20,480 floats

    gemm_ab<<<dim3(80, 2), 32, 0, stream>>>(S, W1, AB);
    build_x1<<<(ROWS * FEAT + 255) / 256, 256, 0, stream>>>(AB, b1, X1);
    gemm_x2<<<dim3(40, 62), 32, 0, stream>>>(X1, W2, b2, X2);
    ctx_c2<<<(WAY * FEAT + 255) / 256, 256, 0, stream>>>(X2, C2);
    dist_kernel<<<(NQ * WAY) / 8, 256, 0, stream>>>(S, Q, C2, out);
}